// DMS_STAttention_48722109006179
// MI455X (gfx1250) — compile-verified
//
#include <hip/hip_runtime.h>
#include <cstdint>

// ---------------------------------------------------------------------------
// DMS_STAttention — algebraic collapse:
// All head-projection vectors have H == 1, so _attention_layer ends with
// softmax over a singleton axis -> exactly 1.0 everywhere, independent of the
// GEMMs / edge scatter / LeakyReLU. Hence:
//     sa[b,t,i,j,0] = 1.0f + sa_bias[t,i,j]
//     ta[b,j,s,t,0] = 1.0f + ta_bias[j,s,t]
// Pure broadcast: read 28 KB of bias, write 57.7 MB  ->  store-BW bound,
// ~2.5 us floor at 23.3 TB/s. Bias is staged in LDS via the gfx1250 async
// global->LDS path (ASYNCcnt + s_wait_asynccnt); output streamed as b128.
// ---------------------------------------------------------------------------

namespace {
constexpr int T_DIM   = 10;
constexpr int J_DIM   = 22;
constexpr int C_DIM   = 64;
constexpr int SA_N    = T_DIM * J_DIM * J_DIM;   // 4840 floats per batch
constexpr int TA_N    = J_DIM * T_DIM * T_DIM;   // 2200 floats per batch
constexpr int SA_V4   = SA_N / 4;                // 1210 float4 chunks
constexpr int TA_V4   = TA_N / 4;                // 550  float4 chunks
constexpr int NB      = 4;                       // batches per block
constexpr int THREADS = 256;                     // 8 wave32 per WG
}

__global__ __launch_bounds__(THREADS)
void dms_stattention_broadcast(const float* __restrict__ sa_bias,
                               const float* __restrict__ ta_bias,
                               float* __restrict__ out, int B)
{
    __shared__ __align__(16) float smem[SA_N + TA_N];
    const int tid = threadIdx.x;

    // ---- stage biases into LDS with gfx1250 async global->LDS copies ------
    // (VDST = 32-bit LDS byte offset; generic shared pointer truncated to
    //  32 bits is exactly the wave-relative LDS address per the ISA aperture
    //  layout.)
    for (int k = tid; k < SA_V4; k += THREADS) {
        const float* g = sa_bias + 4 * k;
        uint32_t loff = (uint32_t)(uintptr_t)&smem[4 * k];
        asm volatile("global_load_async_to_lds_b128 %0, %1, off"
                     :: "v"(loff), "v"(g) : "memory");
    }
    for (int k = tid; k < TA_V4; k += THREADS) {
        const float* g = ta_bias + 4 * k;
        uint32_t loff = (uint32_t)(uintptr_t)&smem[SA_N + 4 * k];
        asm volatile("global_load_async_to_lds_b128 %0, %1, off"
                     :: "v"(loff), "v"(g) : "memory");
    }
    asm volatile("s_wait_asynccnt 0" ::: "memory");
    __syncthreads();

    // ---- fold in the softmax result (== 1.0) once -------------------------
    for (int i = tid; i < SA_N + TA_N; i += THREADS) smem[i] += 1.0f;
    __syncthreads();

    // ---- stream out: b128 stores, NB batches per block --------------------
    const float4* s4     = (const float4*)smem;
    float4*       out_sa = (float4*)out;
    float4*       out_ta = (float4*)(out + (size_t)B * SA_N);

    const int b0 = blockIdx.x * NB;
#pragma unroll
    for (int bb = 0; bb < NB; ++bb) {
        const int b = b0 + bb;
        if (b >= B) break;
        float4* osa = out_sa + (size_t)b * SA_V4;
        for (int k = tid; k < SA_V4; k += THREADS) osa[k] = s4[k];
        float4* ota = out_ta + (size_t)b * TA_V4;
        for (int k = tid; k < TA_V4; k += THREADS) ota[k] = s4[SA_V4 + k];
    }
}

extern "C" void kernel_launch(void* const* d_in, const int* in_sizes, int n_in,
                              void* d_out, int out_size, void* d_ws, size_t ws_size,
                              hipStream_t stream) {
    // setup_inputs() order: 0=src, 1..10 = weights/head vectors (all dead),
    // 11=sa_bias [T,J,J], 12=ta_bias [J,T,T]
    const float* sa_bias = (const float*)d_in[11];
    const float* ta_bias = (const float*)d_in[12];
    float* out = (float*)d_out;

    const int B    = in_sizes[0] / (C_DIM * T_DIM * J_DIM);  // 2048
    const int grid = (B + NB - 1) / NB;                      // 512 blocks
    dms_stattention_broadcast<<<grid, THREADS, 0, stream>>>(sa_bias, ta_bias,
                                                            out, B);
    (void)n_in; (void)out_size; (void)d_ws; (void)ws_size;
}